// LineageTracerLoss_2851858284964
// MI455X (gfx1250) — compile-verified
//
#include <hip/hip_runtime.h>

// Batch-hard triplet loss, n=8192, d=64, fp32.
// Gram matrix via V_WMMA_F32_16X16X4_F32 (full fp32 precision on the matrix pipe).

typedef __attribute__((ext_vector_type(2))) float v2f;
typedef __attribute__((ext_vector_type(8))) float v8f;

#define TL_MARGIN 0.2f
#define TL_EPS    1e-12f
#define TL_D      64
#define TL_NMAX   8192

__global__ void tl_sqnorm_kernel(const float* __restrict__ E,
                                 float* __restrict__ sq, int n) {
  int i = blockIdx.x * blockDim.x + threadIdx.x;
  if (i >= n) return;
  const float4* row = (const float4*)(E + (size_t)i * TL_D);
  float s = 0.f;
#pragma unroll
  for (int k = 0; k < TL_D / 4; ++k) {
    float4 v = row[k];
    s = fmaf(v.x, v.x, s); s = fmaf(v.y, v.y, s);
    s = fmaf(v.z, v.z, s); s = fmaf(v.w, v.w, s);
  }
  sq[i] = s;
}

__launch_bounds__(256)
__global__ void tl_tile_kernel(const float* __restrict__ E,
                               const int*   __restrict__ T,
                               const float* __restrict__ sq,
                               float* __restrict__ partial,
                               int n) {
  __shared__ float s_sq[TL_NMAX];     // 32 KB
  __shared__ short s_tg[TL_NMAX];     // 16 KB (class ids < 1024)
  __shared__ int   s_ap[16];
  __shared__ int   s_an[16];
  __shared__ float s_term[16];

  const int tid  = threadIdx.x;
  const int wave = tid >> 5;
  const int lane = tid & 31;
  const int l16  = lane & 15;
  const int hi   = lane >> 4;         // 0: lanes 0-15, 1: lanes 16-31
  const int rowBase = blockIdx.x * 16;

  // Stage squared norms + targets into LDS (hot in epilogue).
  for (int i = tid; i < n; i += 256) {
    s_sq[i] = sq[i];
    s_tg[i] = (short)T[i];
  }
  if (tid < 16) {
    s_ap[tid] = 0;                    // bits of 0.0f; every row has self as positive
    s_an[tid] = 0x7F800000;           // bits of +inf
  }
  __syncthreads();

  // A-fragment layout for V_WMMA_F32_16X16X4_F32 (k-step k covers K=4k..4k+3):
  //   lane L in 0-15:  M = L,    holds K = 4k+{0,1}
  //   lane L in 16-31: M = L-16, holds K = 4k+{2,3}
  const float* arow = E + (size_t)(rowBase + l16) * TL_D + 2 * hi;
  v2f a[TL_D / 4];
#pragma unroll
  for (int k = 0; k < TL_D / 4; ++k) a[k] = *(const v2f*)(arow + 4 * k);

  float apmax[8], anmin[8];
  int   rtg[8];
  float rsq[8];
#pragma unroll
  for (int v = 0; v < 8; ++v) {
    apmax[v] = 0.0f;
    anmin[v] = __int_as_float(0x7F800000);
    int r = rowBase + v + 8 * hi;     // C-tile row this lane holds in VGPR v
    rtg[v] = (int)s_tg[r];
    rsq[v] = s_sq[r];
  }

  const int tiles = n >> 4;
  // Uniform per-wave loop: all 32 lanes active at every WMMA (EXEC all-ones).
  for (int t = wave; t < tiles; t += 8) {
    const int colBase = t << 4;
    // B-fragment (4x16): lane L holds N=L&15, K = 4k+{0,1} (lo) / 4k+{2,3} (hi);
    // B[k][n] = E[colBase+n][k]  =>  same access pattern as A, column-row base.
    const float* brow = E + (size_t)(colBase + l16) * TL_D + 2 * hi;
    v8f c = {0.f, 0.f, 0.f, 0.f, 0.f, 0.f, 0.f, 0.f};
#pragma unroll
    for (int k = 0; k < TL_D / 4; ++k) {
      v2f b = *(const v2f*)(brow + 4 * k);
      c = __builtin_amdgcn_wmma_f32_16x16x4_f32(false, a[k], false, b,
                                                (short)0, c, false, false);
    }
    const int   colC = colBase + l16;
    const float sqc  = s_sq[colC];
    const int   tgc  = (int)s_tg[colC];
#pragma unroll
    for (int v = 0; v < 8; ++v) {
      float d2   = rsq[v] + sqc - 2.0f * c[v];
      float dist = __builtin_sqrtf(fmaxf(d2, TL_EPS));
      if (tgc == rtg[v]) apmax[v] = fmaxf(apmax[v], dist);
      else               anmin[v] = fminf(anmin[v], dist);
    }
  }

  // Merge lane-local extrema: all distances >= sqrt(EPS) > 0, so float order
  // == signed-int order on the raw bits -> LDS integer atomic max/min is exact.
#pragma unroll
  for (int v = 0; v < 8; ++v) {
    int slot = v + 8 * hi;
    atomicMax(&s_ap[slot], __float_as_int(apmax[v]));
    atomicMin(&s_an[slot], __float_as_int(anmin[v]));
  }
  __syncthreads();

  if (tid < 16) {
    float ap = __int_as_float(s_ap[tid]);
    float an = __int_as_float(s_an[tid]);
    s_term[tid] = fmaxf(ap - an + TL_MARGIN, 0.0f);  // relu(ap - an + margin)
  }
  __syncthreads();
  if (tid == 0) {
    float s = 0.f;
#pragma unroll
    for (int i = 0; i < 16; ++i) s += s_term[i];
    partial[blockIdx.x] = s;                          // deterministic per-block sum
  }
}

__global__ void tl_finalize_kernel(const float* __restrict__ partial, int m,
                                   float* __restrict__ out, float invn) {
  __shared__ float s[256];
  float acc = 0.f;
  for (int i = threadIdx.x; i < m; i += 256) acc += partial[i];
  s[threadIdx.x] = acc;
  __syncthreads();
  for (int off = 128; off > 0; off >>= 1) {
    if ((int)threadIdx.x < off) s[threadIdx.x] += s[threadIdx.x + off];
    __syncthreads();
  }
  if (threadIdx.x == 0) out[0] = s[0] * invn;         // mean
}

extern "C" void kernel_launch(void* const* d_in, const int* in_sizes, int n_in,
                              void* d_out, int out_size, void* d_ws, size_t ws_size,
                              hipStream_t stream) {
  const float* E  = (const float*)d_in[0];   // embeddings [n, 64] fp32
  const int*   T  = (const int*)d_in[1];     // targets [n] (integer per harness contract)
  float*       out = (float*)d_out;          // loss [1] fp32

  const int n = in_sizes[0] / TL_D;          // 8192
  const int rowBlocks = n / 16;              // 512

  float* sq      = (float*)d_ws;             // [n] squared norms
  float* partial = sq + n;                   // [rowBlocks] per-block loss sums

  tl_sqnorm_kernel<<<(n + 255) / 256, 256, 0, stream>>>(E, sq, n);
  tl_tile_kernel<<<rowBlocks, 256, 0, stream>>>(E, T, sq, partial, n);
  tl_finalize_kernel<<<1, 256, 0, stream>>>(partial, rowBlocks, out, 1.0f / (float)n);
}